// Base_CGCNN_Module_19009525252275
// MI455X (gfx1250) — compile-verified
//
#include <hip/hip_runtime.h>

#define FEA   128
#define WIN   16    // segments per block = WMMA M dimension

typedef __attribute__((ext_vector_type(2))) float v2f;
typedef __attribute__((ext_vector_type(4))) float v4f;
typedef __attribute__((ext_vector_type(8))) float v8f;
typedef __attribute__((ext_vector_type(4))) int   v4i;

__device__ __forceinline__ int lowerBound(const int* __restrict__ a, int n, int key) {
    int lo = 0, hi = n;
    while (lo < hi) { int mid = (lo + hi) >> 1; if (a[mid] < key) lo = mid + 1; else hi = mid; }
    return lo;
}

__device__ __forceinline__ int imin(int a, int b) { return a < b ? a : b; }

// ---------------------------------------------------------------------------
// Fused segment-mean, one kernel, no atomics, deterministic.
//
// Block b owns segments [16b, 16b+16) -> fixed WMMA window s_base = 16b.
// readfirstlane-pinned binary searches give the block's atom range [aBeg,aEnd)
// as SGPRs, and the per-wave chunk bounds are ALSO pinned (q is wave-uniform),
// so the hot loop is fully scalar: s_cmp/s_cbranch control flow, scalar-path
// seg-id quads, vector loads only for B data.
// 8 waves = 4 atom-quarters x 2 feature-halves. Each wave accumulates
//   D[m][col] += sum_k A[m][k]*B[k][col]
// with V_WMMA_F32_16X16X4_F32, A = one-hot (seg(atom)==s_base+m): edge-chunk
// atoms from neighboring windows fail the compare -> no masking needed.
// Interleaved feature mapping (tile j, col n <-> feature 64*fh + 4n + j)
// makes each lane's B data for all 4 tiles one contiguous float4 per K-row.
// Two chunks per iteration (x unroll 2) keep ~4 KB of b128 loads in flight
// per wave for latency coverage at full HBM bandwidth.
// Partial sums combine deterministically via LDS tiles, get divided by
// binary-searched counts, and are stored with b128 stores.
// ---------------------------------------------------------------------------
__global__ void __launch_bounds__(256)
segmean_wmma(const float* __restrict__ fea, const int* __restrict__ ids,
             float* __restrict__ out, int n, int n0) {
    __shared__ v4f   tiles[8 * WIN * 16];   // [wave][row 0..15][feat4 0..15]  (32 KB)
    __shared__ float sInv[WIN];

    const int s_base = blockIdx.x * WIN;
    const int lane   = threadIdx.x & 31;
    const int w      = threadIdx.x >> 5;    // wave 0..7 (wave-uniform)
    const int q      = w >> 1;              // atom quarter 0..3 (wave-uniform)
    const int fh     = w & 1;               // feature half 0..1 (wave-uniform)
    const int half   = lane >> 4;           // A/C: lanes 16-31 hold K=2,3 / M=8..15
    const int m      = lane & 15;           // A/C row (segment offset in window)
    const int nf     = lane & 15;           // B/C column index
    const int fb     = fh * 64 + 4 * nf;    // lane's feature base (16B aligned)
    const int basem  = s_base + m;          // one-hot compare value

    // ---- block atom range; pin to SGPRs ----
    const int aBeg = __builtin_amdgcn_readfirstlane(lowerBound(ids, n, s_base));
    const int aEnd = __builtin_amdgcn_readfirstlane(lowerBound(ids, n, s_base + WIN));

    // ---- per-lane segment count (lane nf handles segment s_base+nf) ----
    {
        const int sL = s_base + nf;
        int lo = aBeg, hi = aEnd;
        while (lo < hi) { int mid = (lo + hi) >> 1; if (ids[mid] <  sL) lo = mid + 1; else hi = mid; }
        const int lb = lo; hi = aEnd;
        while (lo < hi) { int mid = (lo + hi) >> 1; if (ids[mid] <= sL) lo = mid + 1; else hi = mid; }
        const int cnt = lo - lb;
        if (threadIdx.x < WIN) sInv[threadIdx.x] = 1.0f / (float)(cnt < 1 ? 1 : cnt);
    }

    // ---- chunk range, split into 4 quarters; pin per-wave bounds to SGPRs ----
    const int cBeg  = aBeg >> 2;
    const int cEnd  = (aEnd + 3) >> 2;
    const int cSafe = imin(cEnd, n >> 2);       // chunks fully in-bounds
    const int lenq  = (cSafe - cBeg + 3) >> 2;
    const int myBeg = __builtin_amdgcn_readfirstlane(cBeg + q * lenq);
    const int myEnd = __builtin_amdgcn_readfirstlane(imin(myBeg + lenq, cSafe));

    v8f c0 = {}, c1 = {}, c2 = {}, c3 = {};

    // compute phase for one chunk (loads are done by the caller, up front)
    auto mm = [&](v4i sq, v4f bx, v4f by) {
        const int va0 = half ? sq[2] : sq[0];   // seg of K = 2*half
        const int va1 = half ? sq[3] : sq[1];   // seg of K = 2*half+1
        const float ax = (va0 == basem) ? 1.0f : 0.0f;
        const float ay = (va1 == basem) ? 1.0f : 0.0f;
        const v2f a = {ax, ay};
        c0 = __builtin_amdgcn_wmma_f32_16x16x4_f32(false, a, false, (v2f){bx[0], by[0]}, (short)0, c0, false, false);
        c1 = __builtin_amdgcn_wmma_f32_16x16x4_f32(false, a, false, (v2f){bx[1], by[1]}, (short)0, c1, false, false);
        c2 = __builtin_amdgcn_wmma_f32_16x16x4_f32(false, a, false, (v2f){bx[2], by[2]}, (short)0, c2, false, false);
        c3 = __builtin_amdgcn_wmma_f32_16x16x4_f32(false, a, false, (v2f){bx[3], by[3]}, (short)0, c3, false, false);
    };

    int c = myBeg;
    #pragma unroll 2
    for (; c + 2 <= myEnd; c += 2) {
        // ---- issue all loads for both chunks first (one clause, 6 loads) ----
        const v4i sqA = *(const v4i*)(ids + 4 * c);        // uniform -> scalar path
        const v4i sqB = *(const v4i*)(ids + 4 * c + 4);
        const float* rp = fea + ((long)(4 * c) + 2 * half) * FEA + fb;
        const v4f bxA = *(const v4f*)rp;                   // chunk c,   K = 2*half
        const v4f byA = *(const v4f*)(rp + FEA);           // chunk c,   K = 2*half+1
        const v4f bxB = *(const v4f*)(rp + 4 * FEA);       // chunk c+1, K = 2*half
        const v4f byB = *(const v4f*)(rp + 5 * FEA);       // chunk c+1, K = 2*half+1
        // ---- then compute ----
        mm(sqA, bxA, byA);
        mm(sqB, bxB, byB);
    }
    if (c < myEnd) {                                       // odd remainder chunk
        const v4i sq = *(const v4i*)(ids + 4 * c);
        const float* rp = fea + ((long)(4 * c) + 2 * half) * FEA + fb;
        const v4f bx = *(const v4f*)rp;
        const v4f by = *(const v4f*)(rp + FEA);
        mm(sq, bx, by);
    }

    // ---- epilogue: partial last chunk (only when n % 4 != 0, last block) ----
    if (q == 3 && cEnd > cSafe) {
        const int g0 = 4 * cSafe;
        #pragma unroll
        for (int k = 0; k < 4; ++k) {
            const int atom = g0 + k;
            if (atom < n) {
                const int sv = ids[atom];
                const v4f bz = *(const v4f*)(fea + (long)atom * FEA + fb);
                const float ax = ((k == 2 * half)     && sv == basem) ? 1.0f : 0.0f;
                const float ay = ((k == 2 * half + 1) && sv == basem) ? 1.0f : 0.0f;
                const v2f a = {ax, ay};
                c0 = __builtin_amdgcn_wmma_f32_16x16x4_f32(false, a, false, (v2f){bz[0], bz[0]}, (short)0, c0, false, false);
                c1 = __builtin_amdgcn_wmma_f32_16x16x4_f32(false, a, false, (v2f){bz[1], bz[1]}, (short)0, c1, false, false);
                c2 = __builtin_amdgcn_wmma_f32_16x16x4_f32(false, a, false, (v2f){bz[2], bz[2]}, (short)0, c2, false, false);
                c3 = __builtin_amdgcn_wmma_f32_16x16x4_f32(false, a, false, (v2f){bz[3], bz[3]}, (short)0, c3, false, false);
            }
        }
    }

    // ---- stash this wave's 16x64 partial tile in LDS ----
    #pragma unroll
    for (int r = 0; r < 8; ++r) {
        const int row = r + half * 8;                       // C: vgpr r -> M = r (+8)
        tiles[(w * WIN + row) * 16 + nf] = (v4f){c0[r], c1[r], c2[r], c3[r]};
    }
    __syncthreads();

    // ---- deterministic combine over the 4 atom-quarters, divide, store ----
    // output viewed as [WIN rows][32 float4 columns]; 512 slots over 256 threads
    #pragma unroll
    for (int slot = threadIdx.x; slot < WIN * 32; slot += 256) {
        const int row  = slot >> 5;
        const int col4 = slot & 31;
        const int fh2  = col4 >> 4;
        const int c4   = col4 & 15;
        v4f acc = tiles[((0 * 2 + fh2) * WIN + row) * 16 + c4];
        acc    += tiles[((1 * 2 + fh2) * WIN + row) * 16 + c4];
        acc    += tiles[((2 * 2 + fh2) * WIN + row) * 16 + c4];
        acc    += tiles[((3 * 2 + fh2) * WIN + row) * 16 + c4];
        const int s = s_base + row;
        if (s < n0) {
            const float iv = sInv[row];
            *(v4f*)(out + (long)s * FEA + col4 * 4) = acc * iv;
        }
    }
}

// ---------------------------------------------------------------------------
extern "C" void kernel_launch(void* const* d_in, const int* in_sizes, int n_in,
                              void* d_out, int out_size, void* d_ws, size_t ws_size,
                              hipStream_t stream) {
    const float* fea = (const float*)d_in[0];
    const int*   ids = (const int*)  d_in[1];
    float*       out = (float*)      d_out;

    const int n  = in_sizes[1];          // number of atoms
    const int n0 = out_size / FEA;       // number of segments

    const int blocks = (n0 + WIN - 1) / WIN;
    segmean_wmma<<<blocks, 256, 0, stream>>>(fea, ids, out, n, n0);

    (void)d_ws; (void)ws_size; (void)n_in;
}